// AffineFakeQuantizer_10771777978901
// MI455X (gfx1250) — compile-verified
//
#include <hip/hip_runtime.h>
#include <stdint.h>

// ---------------------------------------------------------------------------
// MX-FP4 (E2M1, block=16) fake quantizer with FP8-E4M3 block scales.
// Memory-bound: 805 MB total traffic -> ~35us floor @ 23.3 TB/s.
// CDNA5 paths: 8-deep global_load_async_to_lds_b128 pipeline where the shared
// 24-bit IOFFSET advances BOTH the global stream and the LDS ring (same 4096B
// stride), s_wait_asynccnt draining, ds_swizzle wave32 butterflies, NT stores.
// ---------------------------------------------------------------------------

#define TPB    256
#define ITERS  8
#define F32MAX 3.402823466e38f

typedef float v4f __attribute__((ext_vector_type(4)));

// The async-to-LDS builtin takes pointers to int4 vectors in AS1/AS3.
typedef int v4i_vs __attribute__((__vector_size__(16)));
typedef __attribute__((address_space(1))) v4i_vs* as1_v4i_p;
typedef __attribute__((address_space(3))) v4i_vs* as3_v4i_p;

#if defined(__has_builtin)
#  if __has_builtin(__builtin_amdgcn_global_load_async_to_lds_b128) && \
      __has_builtin(__builtin_amdgcn_s_wait_asynccnt)
#    define USE_ASYNC 1
#  endif
#endif
#ifndef USE_ASYNC
#  define USE_ASYNC 0
#endif

// wave32 group-of-4 butterfly: xor=1 -> 0x041F, xor=2 -> 0x081F
#define SWZ_F(x, imm) \
  __int_as_float(__builtin_amdgcn_ds_swizzle(__float_as_int(x), (imm)))

// f32 -> float8_e4m3fn -> f32, RNE, branchless (scale diverges per 4 lanes).
__device__ __forceinline__ float e4m3_roundtrip(float s) {
  float sub = rintf(s * 512.0f) * 0.001953125f;   // subnormal quantum 2^-9
  uint32_t u = __float_as_uint(s);
  u += 0x0007FFFFu + ((u >> 20) & 1u);            // RNE to 3 mantissa bits
  u &= 0xFFF00000u;
  float nrm = __uint_as_float(u);
  return (s < 0.015625f) ? sub : nrm;
}

// Fake-quantize to E2M1 grid {0,.5,1,1.5,2,3,4,6} (signed).  The reference's
// subnormal code clamp (a in [0.75,1) -> 0.5) folds into qmax = a<1 ? .5 : 6.
__device__ __forceinline__ float quant_e2m1(float xq) {
  float a  = __builtin_fabsf(xq);
  int   eb = (int)(__float_as_uint(a) >> 23);     // biased exponent
  int   k  = (eb > 127) ? (eb - 128) : -1;        // step = 2^k
  float c  = rintf(ldexpf(a, -k));                // v_ldexp + v_rndne (RNE)
  float qmax = (a < 1.0f) ? 0.5f : 6.0f;
  float q  = fminf(ldexpf(c, k), qmax);
  return __builtin_copysignf(q, xq);
}

__device__ __forceinline__ float fix_special(float x, float o, float s) {
  if (x ==  __builtin_inff()) return  F32MAX * s;
  if (x == -__builtin_inff()) return -F32MAX * s;
  if (x != x)                 return __builtin_nanf("");
  return o;
}

// One float4 per lane; 4 consecutive lanes form one 16-element quant block.
__device__ __forceinline__ v4f process4(v4f x) {
  // block amax of sanitized |x|: v_max_num drops NaN (->0 via max with 0),
  // v_min_num clamps Inf to F32_MAX.  Abs is a free source modifier.
  float m = fmaxf(fmaxf(__builtin_fabsf(x[0]), __builtin_fabsf(x[1])),
                  fmaxf(__builtin_fabsf(x[2]), __builtin_fabsf(x[3])));
  m = fminf(fmaxf(m, 0.0f), F32MAX);
  m = fmaxf(m, SWZ_F(m, 0x041F));                 // xor 1
  m = fmaxf(m, SWZ_F(m, 0x081F));                 // xor 2 -> max over 16 elems

  float s   = e4m3_roundtrip(fminf(m / 6.0f, 448.0f));
  float inv = 1.0f / (s + 1e-8f);

  v4f o;
  o[0] = quant_e2m1(x[0] * inv) * s;
  o[1] = quant_e2m1(x[1] * inv) * s;
  o[2] = quant_e2m1(x[2] * inv) * s;
  o[3] = quant_e2m1(x[3] * inv) * s;

  // rare NaN/Inf overrides (branch almost always uniformly not-taken)
  bool sp = !(__builtin_fabsf(x[0]) <= F32MAX) ||
            !(__builtin_fabsf(x[1]) <= F32MAX) ||
            !(__builtin_fabsf(x[2]) <= F32MAX) ||
            !(__builtin_fabsf(x[3]) <= F32MAX);
  if (sp) {
    o[0] = fix_special(x[0], o[0], s);
    o[1] = fix_special(x[1], o[1], s);
    o[2] = fix_special(x[2], o[2], s);
    o[3] = fix_special(x[3], o[3], s);
  }
  return o;
}

// Fast path: grid exactly covers n4 (no guards, no tail).
__global__ __launch_bounds__(TPB) void fq4_kernel(const float* __restrict__ in,
                                                  float* __restrict__ out) {
  const int  tid  = threadIdx.x;
  const long base = (long)blockIdx.x * (TPB * (long)ITERS) + tid;
  const v4f* __restrict__ pg = (const v4f*)in + base;   // per-lane global src
  v4f*       __restrict__ po = (v4f*)out + base;        // per-lane global dst

#if USE_ASYNC && defined(__HIP_DEVICE_COMPILE__)
  // 8-slot LDS ring, row stride 4096B == global per-iteration stride, so the
  // instruction IOFFSET (applied by HW to BOTH addresses) does all stepping.
  __shared__ v4f stage[ITERS][TPB];                     // 32 KB
  const uint32_t lds0 = (uint32_t)(uintptr_t)(&stage[0][tid]);

#define ISSUE(IT)                                                        \
  __builtin_amdgcn_global_load_async_to_lds_b128(                        \
      (as1_v4i_p)(uintptr_t)pg, (as3_v4i_p)(uintptr_t)lds0,              \
      (IT) * TPB * 16, 0)

  ISSUE(0); ISSUE(1); ISSUE(2); ISSUE(3);
  ISSUE(4); ISSUE(5); ISSUE(6); ISSUE(7);

#define STEP(IT)                                                         \
  do {                                                                   \
    __builtin_amdgcn_s_wait_asynccnt(ITERS - 1 - (IT));                  \
    v4f x_ = stage[IT][tid];                    /* ds_load_b128 */       \
    __builtin_nontemporal_store(process4(x_), po + (IT) * TPB);          \
  } while (0)

  STEP(0); STEP(1); STEP(2); STEP(3);
  STEP(4); STEP(5); STEP(6); STEP(7);
#undef ISSUE
#undef STEP
#else
#pragma unroll
  for (int it = 0; it < ITERS; ++it) {
    v4f x = __builtin_nontemporal_load(pg + it * TPB);
    __builtin_nontemporal_store(process4(x), po + it * TPB);
  }
#endif
}

// Guarded fallback (only used if n4 is not a multiple of TPB*ITERS).
__global__ __launch_bounds__(TPB) void fq4_kernel_tail(const float* __restrict__ in,
                                                       float* __restrict__ out,
                                                       long n4) {
  const long idx = (long)blockIdx.x * TPB + threadIdx.x;
  if (idx < n4) {
    v4f x = __builtin_nontemporal_load((const v4f*)in + idx);
    __builtin_nontemporal_store(process4(x), (v4f*)out + idx);
  }
}

extern "C" void kernel_launch(void* const* d_in, const int* in_sizes, int n_in,
                              void* d_out, int out_size, void* d_ws, size_t ws_size,
                              hipStream_t stream) {
  (void)n_in; (void)out_size; (void)d_ws; (void)ws_size;
  const float* x   = (const float*)d_in[0];
  float*       out = (float*)d_out;
  long n  = (long)in_sizes[0];                  // 8*4096*3072, divisible by 16
  long n4 = n / 4;
  long per_block = (long)TPB * ITERS;
  if (n4 % per_block == 0) {
    int blocks = (int)(n4 / per_block);         // 12288 for the reference shape
    fq4_kernel<<<blocks, TPB, 0, stream>>>(x, out);
  } else {
    int blocks = (int)((n4 + TPB - 1) / TPB);
    fq4_kernel_tail<<<blocks, TPB, 0, stream>>>(x, out, n4);
  }
}